// HyperbubbleGNN_41652592837293
// MI455X (gfx1250) — compile-verified
//
#include <hip/hip_runtime.h>
#include <hip/hip_bf16.h>

typedef __attribute__((ext_vector_type(2))) float v2f;
typedef __attribute__((ext_vector_type(8))) float v8f;

#define EMBED 16
#define HID 16
#define NODE_IN 17        // EMBED + 1
#define NODE_PAD 20       // padded X stride (multiple of 4)
#define EFEAT 5
#define EMLP_K 37         // 2*HID + EFEAT
#define EMLP_PAD 40       // padded edge-feature stride
#define LSEQ 64
#define EPB 128           // edges per block in edge MLP (8 waves * 16)

// ds_swizzle group-of-32 butterfly: xor_mask<<10 | or 0<<5 | and 0x1f
template<int XORM>
__device__ __forceinline__ float hb_swz_add(float x) {
    constexpr int pat = (XORM << 10) | 0x1f;
    int v = __builtin_amdgcn_ds_swizzle(__float_as_int(x), pat);
    return x + __int_as_float(v);
}

// ---------------------------------------------------------------------------
// 1) encode_nodes: masked mean of token embeddings + covariate -> X [N,20]
// ---------------------------------------------------------------------------
__global__ void hb_encode(const int* __restrict__ tok,
                          const float* __restrict__ xcov,
                          const float* __restrict__ embed,
                          float* __restrict__ X, int n) {
    __shared__ float emb[6 * EMBED];
    if (threadIdx.x < 6 * EMBED) emb[threadIdx.x] = embed[threadIdx.x];
    __syncthreads();
    int i = blockIdx.x * blockDim.x + threadIdx.x;
    if (i >= n) return;
    float acc[EMBED];
#pragma unroll
    for (int d = 0; d < EMBED; d++) acc[d] = 0.f;
    float cnt = 0.f;
    const int4* t4 = (const int4*)(tok + (long)i * LSEQ);
#pragma unroll 4
    for (int q = 0; q < LSEQ / 4; q++) {
        int4 v4 = t4[q];                        // global_load_b128
        int vv[4] = {v4.x, v4.y, v4.z, v4.w};
#pragma unroll
        for (int j = 0; j < 4; j++) {
            int v = vv[j];
            if (v != 0) {
                cnt += 1.f;
                const float* er = &emb[v * EMBED];
#pragma unroll
                for (int d = 0; d < EMBED; d++) acc[d] += er[d];
            }
        }
    }
    float inv = 1.f / fmaxf(cnt, 1.f);
    float* xr = X + (long)i * NODE_PAD;
#pragma unroll
    for (int d = 0; d < EMBED; d++) xr[d] = acc[d] * inv;
    xr[16] = xcov[i];
    xr[17] = 0.f; xr[18] = 0.f; xr[19] = 0.f;
}

// ---------------------------------------------------------------------------
// 2) pad weights: w1[17x16] -> w1p[20x16], we1[37x16] -> we1p[40x16]
// ---------------------------------------------------------------------------
__global__ void hb_pad_weights(const float* __restrict__ w1,
                               const float* __restrict__ we1,
                               float* __restrict__ w1p, float* __restrict__ we1p) {
    for (int i = threadIdx.x; i < NODE_PAD * HID; i += blockDim.x)
        w1p[i] = (i < NODE_IN * HID) ? w1[i] : 0.f;
    for (int i = threadIdx.x; i < EMLP_PAD * HID; i += blockDim.x)
        we1p[i] = (i < EMLP_K * HID) ? we1[i] : 0.f;
}

// ---------------------------------------------------------------------------
// 3) degree: zero, atomic count of valid (same-graph) out-edges, rsqrt
// ---------------------------------------------------------------------------
__global__ void hb_zero(float* __restrict__ p, long n) {
    long i = (long)blockIdx.x * blockDim.x + threadIdx.x;
    if (i < n) p[i] = 0.f;
}

__global__ void hb_deg_count(const int* __restrict__ src, const int* __restrict__ dst,
                             const int* __restrict__ batch, float* __restrict__ deg, int E) {
    int e = blockIdx.x * blockDim.x + threadIdx.x;
    if (e >= E) return;
    int s = src[e], d = dst[e];
    if (batch[s] == batch[d]) atomicAdd(&deg[s], 1.0f);
}

__global__ void hb_deg_finalize(float* __restrict__ dinv, int n) {
    int i = blockIdx.x * blockDim.x + threadIdx.x;
    if (i >= n) return;
    dinv[i] = rsqrtf(dinv[i] + 1.0f + 1e-8f);
}

// ---------------------------------------------------------------------------
// 4) WMMA GEMM: Out[N,16] = A[N,4*KC](lda) @ W[4*KC,16] + bias. Wave = 16 rows.
// ---------------------------------------------------------------------------
template<int KC>
__global__ void hb_gemm16(const float* __restrict__ A, int lda,
                          const float* __restrict__ W, const float* __restrict__ bias,
                          float* __restrict__ Out, int nrows) {
    int wave = (blockIdx.x * blockDim.x + threadIdx.x) >> 5;
    int lane = threadIdx.x & 31;
    int row0 = wave * 16;
    if (row0 >= nrows) return;                  // wave-uniform: EXEC stays all-1
    int half = lane >> 4;
    int lr = lane & 15;
    int rowA = row0 + lr; if (rowA >= nrows) rowA = nrows - 1;
    const float* arow = A + (long)rowA * lda;
    v8f c = {};
#pragma unroll
    for (int kc = 0; kc < KC; kc++) {
        int k0 = kc * 4 + half * 2;
        v2f a = *reinterpret_cast<const v2f*>(arow + k0);   // global_load_b64
        v2f b;
        b[0] = W[k0 * HID + lr];
        b[1] = W[k0 * HID + HID + lr];
        c = __builtin_amdgcn_wmma_f32_16x16x4_f32(false, a, false, b, (short)0, c,
                                                  false, false);
    }
    float bv = bias[lr];
#pragma unroll
    for (int i = 0; i < 8; i++) {
        int row = row0 + i + half * 8;
        if (row < nrows) Out[(long)row * HID + lr] = c[i] + bv;
    }
}

// ---------------------------------------------------------------------------
// 5) message passing: agg = dinv*M (self) + scatter_add(valid * dinv[dst]*M[dst])
// ---------------------------------------------------------------------------
__global__ void hb_agg_init(const float* __restrict__ dinv, const float* __restrict__ M,
                            float* __restrict__ agg, int n) {
    long t = (long)blockIdx.x * blockDim.x + threadIdx.x;   // over n*4 float4s
    if (t >= (long)n * (HID / 4)) return;
    int node = (int)(t >> 2);
    float s = dinv[node];
    float4 m = ((const float4*)M)[t];
    float4 r; r.x = s * m.x; r.y = s * m.y; r.z = s * m.z; r.w = s * m.w;
    ((float4*)agg)[t] = r;
}

__global__ void hb_msg(const int* __restrict__ src, const int* __restrict__ dst,
                       const int* __restrict__ batch, const float* __restrict__ dinv,
                       const float* __restrict__ M, float* __restrict__ agg, int E) {
    long t = (long)blockIdx.x * blockDim.x + threadIdx.x;   // over E*4
    if (t >= (long)E * (HID / 4)) return;
    int e = (int)(t >> 2);
    int c4 = (int)(t & 3);
    int s = src[e], d = dst[e];
    if (batch[s] != batch[d]) return;
    float w = dinv[d];
    float4 m = *(const float4*)(M + (long)d * HID + c4 * 4);    // global_load_b128
    float* ap = agg + (long)s * HID + c4 * 4;
    atomicAdd(ap + 0, w * m.x);
    atomicAdd(ap + 1, w * m.y);
    atomicAdd(ap + 2, w * m.z);
    atomicAdd(ap + 3, w * m.w);
}

__global__ void hb_relu_scale(const float* __restrict__ dinv, float* __restrict__ H, int n) {
    long t = (long)blockIdx.x * blockDim.x + threadIdx.x;   // over n*4 float4s
    if (t >= (long)n * (HID / 4)) return;
    int node = (int)(t >> 2);
    float s = dinv[node];
    float4 h = ((const float4*)H)[t];
    float4 r;
    r.x = fmaxf(s * h.x, 0.f); r.y = fmaxf(s * h.y, 0.f);
    r.z = fmaxf(s * h.z, 0.f); r.w = fmaxf(s * h.w, 0.f);
    ((float4*)H)[t] = r;
}

// ---------------------------------------------------------------------------
// 6) edge MLP: logits = relu([H[s],H[d],ea] @ we1 + be1) @ we2 + be2
//    H rows gathered straight into LDS via async global->LDS (ASYNCcnt path).
// ---------------------------------------------------------------------------
__global__ void hb_edge_mlp(const float* __restrict__ H,
                            const int* __restrict__ src, const int* __restrict__ dst,
                            const float* __restrict__ ea,
                            const float* __restrict__ we1p, const float* __restrict__ be1,
                            const float* __restrict__ we2, const float* __restrict__ be2,
                            float* __restrict__ out, int E) {
    __shared__ float feats[EPB * EMLP_PAD];     // 128*40*4 = 20 KB
    __shared__ float wlog[8][16];
    int tid = threadIdx.x;
    int blkE0 = blockIdx.x * EPB;

    // stage H[src] (part 0) / H[dst] (part 1): 2 threads per edge,
    // async gather directly into LDS (no VGPR round-trip)
    {
        int eidx = tid >> 1;
        int part = tid & 1;
        int e = blkE0 + eidx; if (e >= E) e = E - 1;
        int node = part ? dst[e] : src[e];
        const float* gp = H + (long)node * HID;
        unsigned lds_off = (unsigned)(size_t)&feats[eidx * EMLP_PAD + part * HID];
        asm volatile(
            "global_load_async_to_lds_b128 %0, %1, off\n\t"
            "global_load_async_to_lds_b128 %0, %1, off offset:16\n\t"
            "global_load_async_to_lds_b128 %0, %1, off offset:32\n\t"
            "global_load_async_to_lds_b128 %0, %1, off offset:48"
            :
            : "v"(lds_off), "v"(gp)
            : "memory");
    }
    // stage edge_attr + zero pad: 1 thread per edge (overlaps with async gather)
    if (tid < EPB) {
        int e = blkE0 + tid; if (e >= E) e = E - 1;
        const float* eav = ea + (long)e * EFEAT;
        float* fr = &feats[tid * EMLP_PAD + 2 * HID];
        fr[0] = eav[0]; fr[1] = eav[1]; fr[2] = eav[2]; fr[3] = eav[3]; fr[4] = eav[4];
        fr[5] = 0.f; fr[6] = 0.f; fr[7] = 0.f;
    }
    asm volatile("s_wait_asynccnt 0" ::: "memory");
    __syncthreads();

    int wave = tid >> 5;
    int lane = tid & 31;
    int half = lane >> 4;
    int lr = lane & 15;
    int e0 = blkE0 + wave * 16;
    const float* arow = &feats[(wave * 16 + lr) * EMLP_PAD];

    v8f c = {};
#pragma unroll
    for (int kc = 0; kc < EMLP_PAD / 4; kc++) {
        int k0 = kc * 4 + half * 2;
        v2f a = *reinterpret_cast<const v2f*>(arow + k0);   // ds_load_b64
        v2f b;
        b[0] = we1p[k0 * HID + lr];
        b[1] = we1p[k0 * HID + HID + lr];
        c = __builtin_amdgcn_wmma_f32_16x16x4_f32(false, a, false, b, (short)0, c,
                                                  false, false);
    }

    float b1v = be1[lr];
    float w2v = we2[lr];
    float b2v = be2[0];
#pragma unroll
    for (int i = 0; i < 8; i++) {
        float h = fmaxf(c[i] + b1v, 0.f);
        float p = h * w2v;
        p = hb_swz_add<1>(p);                   // ds_swizzle SWAPX1
        p = hb_swz_add<2>(p);                   // SWAPX2
        p = hb_swz_add<4>(p);                   // SWAPX4
        p = hb_swz_add<8>(p);                   // SWAPX8
        if (lr == 0) wlog[wave][i + half * 8] = p + b2v;
    }
    __syncthreads();
    // coalesced store: 16 lanes per wave write 16 consecutive logits
    if (lane < 16) {
        int row = e0 + lane;
        if (row < E) out[row] = wlog[wave][lane];
    }
}

// ---------------------------------------------------------------------------
// launch
// ---------------------------------------------------------------------------
static inline int cdiv(long a, long b) { return (int)((a + b - 1) / b); }

extern "C" void kernel_launch(void* const* d_in, const int* in_sizes, int n_in,
                              void* d_out, int out_size, void* d_ws, size_t ws_size,
                              hipStream_t stream) {
    const int*   seq_tokens = (const int*)  d_in[0];
    const float* x_cov      = (const float*)d_in[1];
    const int*   edge_index = (const int*)  d_in[2];
    const float* edge_attr  = (const float*)d_in[3];
    const int*   batch      = (const int*)  d_in[4];
    const float* embed      = (const float*)d_in[5];
    const float* w1         = (const float*)d_in[6];
    const float* b1         = (const float*)d_in[7];
    const float* w2         = (const float*)d_in[8];
    const float* b2         = (const float*)d_in[9];
    const float* we1        = (const float*)d_in[10];
    const float* be1        = (const float*)d_in[11];
    const float* we2        = (const float*)d_in[12];
    const float* be2        = (const float*)d_in[13];
    float* out = (float*)d_out;

    const int N = in_sizes[1];          // x_cov is [N,1]
    const int E = in_sizes[2] / 2;      // edge_index is [2,E]
    const int* src = edge_index;
    const int* dst = edge_index + E;

    // workspace (floats): dinv[N] | X[20N] | M[16N] | AGG[16N] | w1p[320] | we1p[640]
    float* ws    = (float*)d_ws;
    float* dinv  = ws;
    float* Xbuf  = dinv + N;
    float* Mbuf  = Xbuf + (long)N * NODE_PAD;
    float* Abuf  = Mbuf + (long)N * HID;
    float* w1p   = Abuf + (long)N * HID;
    float* we1p  = w1p + NODE_PAD * HID;
    // layer-2 agg / H2 recycles Xbuf (dead after layer-1 GEMM), stride 16
    float* A2buf = Xbuf;

    const int BLK = 256;

    hb_encode<<<cdiv(N, BLK), BLK, 0, stream>>>(seq_tokens, x_cov, embed, Xbuf, N);
    hb_pad_weights<<<1, BLK, 0, stream>>>(w1, we1, w1p, we1p);

    hb_zero<<<cdiv(N, BLK), BLK, 0, stream>>>(dinv, N);
    hb_deg_count<<<cdiv(E, BLK), BLK, 0, stream>>>(src, dst, batch, dinv, E);
    hb_deg_finalize<<<cdiv(N, BLK), BLK, 0, stream>>>(dinv, N);

    const int gemm_thr_n = cdiv(N, 16) * 32;

    // --- GCN layer 1 (K = 20 padded -> 5 chunks) ---
    hb_gemm16<5><<<cdiv(gemm_thr_n, BLK), BLK, 0, stream>>>(Xbuf, NODE_PAD, w1p, b1,
                                                            Mbuf, N);
    hb_agg_init<<<cdiv((long)N * 4, BLK), BLK, 0, stream>>>(dinv, Mbuf, Abuf, N);
    hb_msg<<<cdiv((long)E * 4, BLK), BLK, 0, stream>>>(src, dst, batch, dinv, Mbuf,
                                                       Abuf, E);
    hb_relu_scale<<<cdiv((long)N * 4, BLK), BLK, 0, stream>>>(dinv, Abuf, N);

    // --- GCN layer 2 (K = 16 -> 4 chunks) ---
    hb_gemm16<4><<<cdiv(gemm_thr_n, BLK), BLK, 0, stream>>>(Abuf, HID, w2, b2,
                                                            Mbuf, N);
    hb_agg_init<<<cdiv((long)N * 4, BLK), BLK, 0, stream>>>(dinv, Mbuf, A2buf, N);
    hb_msg<<<cdiv((long)E * 4, BLK), BLK, 0, stream>>>(src, dst, batch, dinv, Mbuf,
                                                       A2buf, E);
    hb_relu_scale<<<cdiv((long)N * 4, BLK), BLK, 0, stream>>>(dinv, A2buf, N);

    // --- edge MLP ---
    hb_edge_mlp<<<cdiv(E, EPB), BLK, 0, stream>>>(A2buf, src, dst, edge_attr,
                                                  we1p, be1, we2, be2, out, E);
}